// FirstDeriv_6227702579643
// MI455X (gfx1250) — compile-verified
//
#include <hip/hip_runtime.h>
#include <hip/hip_bf16.h>

#define WAVE_SIZE       32
#define WAVES_PER_BLOCK 8
#define BLOCK_SIZE      (WAVE_SIZE * WAVES_PER_BLOCK)
#define K_EDGES         32
#define CHUNK_E         8          // edges staged per chunk
#define ROW_I2          9          // int2 per LDS row: 8 data + 1 pad -> 72B stride

// CDNA5 async global->LDS path (gfx1250 only; device pass only).
#if defined(__gfx1250__) && \
    __has_builtin(__builtin_amdgcn_global_load_async_to_lds_b128) && \
    __has_builtin(__builtin_amdgcn_s_wait_asynccnt)
#define HAVE_ASYNC_LDS 1
#else
#define HAVE_ASYNC_LDS 0
#endif

#if HAVE_ASYNC_LDS
typedef int v4i __attribute__((vector_size(4 * sizeof(int))));
typedef v4i __attribute__((address_space(1)))* v4i_gptr_t;
typedef v4i __attribute__((address_space(3)))* v4i_lptr_t;
#endif

__global__ __launch_bounds__(BLOCK_SIZE) void firstderiv_wls_kernel(
    const float* __restrict__ coords,   // [N,3] f32
    const int*   __restrict__ conn,     // [N,K,2] i32
    const float* __restrict__ y,        // [N,1] f32
    float*       __restrict__ out,      // [3,N] f32 (grads concatenated)
    const int nNodes, const int nTiles)
{
  const int lane = threadIdx.x & (WAVE_SIZE - 1);
  const int wib  = threadIdx.x >> 5;
  const int tile = blockIdx.x * WAVES_PER_BLOCK + wib;
  if (tile >= nTiles) return;                  // wave-uniform

  const int nodeBase = tile * WAVE_SIZE;
  const int myNode   = nodeBase + lane;        // this lane's node
  const bool active  = myNode < nNodes;

  // Per-lane accumulators: A (symmetric 3x3) and rhs.
  float axx = 0.f, axy = 0.f, axz = 0.f, ayy = 0.f, ayz = 0.f, azz = 0.f;
  float r0 = 0.f, r1 = 0.f, r2 = 0.f;

#if HAVE_ASYNC_LDS
  // Double-buffered staging: [wave][buf][node-in-tile][edge-in-chunk] int2,
  // rows padded to 72B so readback banks = (lane*18 + e*2) % 64 are
  // conflict-free across all 32 lanes.
  __shared__ int2 s_idx[WAVES_PER_BLOCK][2][WAVE_SIZE][ROW_I2];

  // Issue one 2KB chunk (32 nodes x 8 edges x 8B) as 4 coalesced b128
  // async copies per lane. 16B element p = j*32+lane -> node m = p>>2,
  // sub-block e2 = p&3 (edges 2*e2..2*e2+1 of the chunk).
  auto issue_chunk = [&](int c, int b) {
    const int k0 = c * CHUNK_E;
    #pragma unroll
    for (int j = 0; j < 4; ++j) {
      const int p  = j * WAVE_SIZE + lane;
      const int m  = p >> 2;
      const int e2 = p & 3;
      int nd = nodeBase + m;
      nd = (nd < nNodes) ? nd : (nNodes - 1);  // clamp: stay in-range
      __builtin_amdgcn_global_load_async_to_lds_b128(
          (v4i_gptr_t)(conn + (size_t)nd * (K_EDGES * 2) + k0 * 2 + e2 * 4),
          (v4i_lptr_t)&s_idx[wib][b][m][e2 * 2],
          0, 0);
    }
  };

  // Consume one staged chunk: 8 edges for this lane's node.
  auto consume_chunk = [&](int b) {
    #pragma unroll
    for (int e = 0; e < CHUNK_E; ++e) {
      const int2 pr = s_idx[wib][b][lane][e];
      const int i1 = pr.x, i2 = pr.y;
      const float dx = coords[3 * i1 + 0] - coords[3 * i2 + 0];
      const float dy = coords[3 * i1 + 1] - coords[3 * i2 + 1];
      const float dz = coords[3 * i1 + 2] - coords[3 * i2 + 2];
      const float du = y[i1] - y[i2];
      const float s2 = dx * dx + dy * dy + dz * dz;
      // Reference: w = 1/dist, inf (dist==0) -> 1. v_rsq_f32 is 1-ULP.
      const float w  = (s2 == 0.0f) ? 1.0f : __builtin_amdgcn_rsqf(s2);
      const float w2 = w * w;
      const float wx = w2 * dx, wy = w2 * dy, wz = w2 * dz;
      axx = fmaf(wx, dx, axx);  axy = fmaf(wx, dy, axy);
      axz = fmaf(wx, dz, axz);  ayy = fmaf(wy, dy, ayy);
      ayz = fmaf(wy, dz, ayz);  azz = fmaf(wz, dz, azz);
      r0  = fmaf(du, wx, r0);   r1  = fmaf(du, wy, r1);
      r2  = fmaf(du, wz, r2);
    }
  };

  // 4-chunk software pipeline, 2 LDS buffers, literal wait immediates.
  issue_chunk(0, 0);
  issue_chunk(1, 1);
  __builtin_amdgcn_s_wait_asynccnt(4);   // chunk 0 done, chunk 1 in flight
  consume_chunk(0);
  issue_chunk(2, 0);
  __builtin_amdgcn_s_wait_asynccnt(4);   // chunk 1 done, chunk 2 in flight
  consume_chunk(1);
  issue_chunk(3, 1);
  __builtin_amdgcn_s_wait_asynccnt(4);   // chunk 2 done, chunk 3 in flight
  consume_chunk(0);
  __builtin_amdgcn_s_wait_asynccnt(0);   // chunk 3 done
  consume_chunk(1);
#else
  // Fallback: direct per-lane loads (256B/lane stream, cache-line reuse).
  {
    const int nd = active ? myNode : (nNodes - 1);
    #pragma unroll 8
    for (int k = 0; k < K_EDGES; ++k) {
      const int2 pr = *(const int2*)(conn + ((size_t)nd * K_EDGES + k) * 2);
      const int i1 = pr.x, i2 = pr.y;
      const float dx = coords[3 * i1 + 0] - coords[3 * i2 + 0];
      const float dy = coords[3 * i1 + 1] - coords[3 * i2 + 1];
      const float dz = coords[3 * i1 + 2] - coords[3 * i2 + 2];
      const float du = y[i1] - y[i2];
      const float s2 = dx * dx + dy * dy + dz * dz;
      const float w  = (s2 == 0.0f) ? 1.0f : __builtin_amdgcn_rsqf(s2);
      const float w2 = w * w;
      const float wx = w2 * dx, wy = w2 * dy, wz = w2 * dz;
      axx = fmaf(wx, dx, axx);  axy = fmaf(wx, dy, axy);
      axz = fmaf(wx, dz, axz);  ayy = fmaf(wy, dy, ayy);
      ayz = fmaf(wy, dz, ayz);  azz = fmaf(wz, dz, azz);
      r0  = fmaf(du, wx, r0);   r1  = fmaf(du, wy, r1);
      r2  = fmaf(du, wz, r2);
    }
  }
#endif

  // Cramer's rule on symmetric 3x3 A with rhs r (per lane = per node).
  const float c0 = ayy * azz - ayz * ayz;
  const float c1 = axy * azz - ayz * axz;
  const float c2 = axy * ayz - ayy * axz;
  const float detA = axx * c0 - axy * c1 + axz * c2;

  const float m1 = r1 * azz - ayz * r2;
  const float m2 = r1 * ayz - ayy * r2;
  const float m3 = axy * r2 - r1 * axz;

  const float d0 = r0 * c0 - axy * m1 + axz * m2;
  const float d1 = axx * m1 - r0 * c1 + axz * m3;
  const float d2 = axx * (ayy * r2 - r1 * ayz) - axy * m3 + r0 * c2;

  // One precise division instead of three (saves ~17 VALU/node).
  const float invDet = 1.0f / detA;

  if (active) {
    out[myNode]              = d0 * invDet;   // coalesced across lanes
    out[nNodes + myNode]     = d1 * invDet;
    out[2 * nNodes + myNode] = d2 * invDet;
  }
}

extern "C" void kernel_launch(void* const* d_in, const int* in_sizes, int n_in,
                              void* d_out, int out_size, void* d_ws, size_t ws_size,
                              hipStream_t stream) {
  const float* coords = (const float*)d_in[0];   // [N,3] f32
  const int*   conn   = (const int*)d_in[1];     // [N,K,2] i32
  const float* y      = (const float*)d_in[2];   // [N,1] f32
  float* out = (float*)d_out;                    // [3,N] f32

  const int nNodes = in_sizes[0] / 3;            // N from coords
  const int nTiles = (nNodes + WAVE_SIZE - 1) / WAVE_SIZE;
  const int grid   = (nTiles + WAVES_PER_BLOCK - 1) / WAVES_PER_BLOCK;

  hipLaunchKernelGGL(firstderiv_wls_kernel, dim3(grid), dim3(BLOCK_SIZE), 0,
                     stream, coords, conn, y, out, nNodes, nTiles);
}